// CoGNN_47562467835947
// MI455X (gfx1250) — compile-verified
//
#include <hip/hip_runtime.h>
#include <hip/hip_bf16.h>

#define N_NODES 50000
#define N_EDGES 640000

typedef __attribute__((ext_vector_type(2))) float v2f;
typedef __attribute__((ext_vector_type(8))) float v8f;

// ---------- helpers ----------
__device__ __forceinline__ unsigned encf(float f) {
    unsigned u = __float_as_uint(f);
    return (u & 0x80000000u) ? ~u : (u | 0x80000000u);
}
__device__ __forceinline__ float decf(unsigned u) {
    return (u & 0x80000000u) ? __uint_as_float(u & 0x7FFFFFFFu) : __uint_as_float(~u);
}
__device__ __forceinline__ float leaky(float v) { return v > 0.f ? v : 0.2f * v; }
__device__ __forceinline__ float wredsum(float v) {
#pragma unroll
    for (int off = 16; off; off >>= 1) v += __shfl_xor(v, off, 32);
    return v;
}

// ---------- f32 WMMA GEMM: OUT[n][o] = sum_k X[n][k]*W[o][k] + bias[o] ----------
// X: N x 128, W: 128 x 128 (row major, row = output feature), OUT: N x 128
// One wave computes a 16x16 tile via 32x v_wmma_f32_16x16x4_f32.
__global__ __launch_bounds__(256) void gemm128_kernel(
    const float* __restrict__ X, const float* __restrict__ W,
    const float* __restrict__ bias, float* __restrict__ OUT) {
    const int wave = threadIdx.x >> 5;      // 0..7 -> column tile
    const int lane = threadIdx.x & 31;
    const int rowTile = blockIdx.x;         // 0..3124 (50000/16)
    const int row  = rowTile * 16 + (lane & 15);
    const int colB = wave * 16;
    const int wrow = colB + (lane & 15);
    const int koff = (lane < 16) ? 0 : 2;   // ISA 32-bit A/B fragment layout
    const float* xp = X + (size_t)row * 128 + koff;
    const float* wp = W + (size_t)wrow * 128 + koff;
    v8f c = {};
#pragma unroll 4
    for (int k = 0; k < 128; k += 4) {
        v2f a = *(const v2f*)(xp + k);  // A[m][k+koff], A[m][k+koff+1]
        v2f b = *(const v2f*)(wp + k);  // B[k+koff][n] = W[n][k+koff], next k
        c = __builtin_amdgcn_wmma_f32_16x16x4_f32(false, a, false, b,
                                                  (short)0, c, false, false);
    }
    const int nIdx  = lane & 15;
    const int mBase = (lane < 16) ? 0 : 8;
    const float bv  = bias[colB + nIdx];
    float* outp = OUT + (size_t)(rowTile * 16) * 128 + colB + nIdx;
#pragma unroll
    for (int v = 0; v < 8; ++v)
        outp[(size_t)(mBase + v) * 128] = c[v] + bv;
}

// ---------- s_rel for all three heads ----------
// consts: [0..1]=srel_in, [2..3]=srel_out, [4..5]=srel_env
__global__ void prep_rel_kernel(
    const float* inWr,  const float* inWrb,  const float* inAw,  const float* inRel,
    const float* outWr, const float* outWrb, const float* outAw, const float* outRel,
    const float* envWr, const float* envWrb, const float* envAw, const float* envRel,
    float* consts) {
    __shared__ float red[128];
    const int t = threadIdx.x; // 0..127
    for (int tt = 0; tt < 2; ++tt) {
        float r = 0.f;
        const float* wrow = envWr + t * 100;
        const float* rel  = envRel + tt * 100;
        for (int j = 0; j < 100; ++j) r += rel[j] * wrow[j];
        r += envWrb[t];
        red[t] = r * envAw[256 + t];
        __syncthreads();
        for (int s = 64; s; s >>= 1) { if (t < s) red[t] += red[t + s]; __syncthreads(); }
        if (t == 0) consts[4 + tt] = red[0];
        __syncthreads();
    }
    if (t == 0) {
        for (int tt = 0; tt < 2; ++tt) {
            float s = 0.f;
            for (int o = 0; o < 2; ++o) {
                float r = 0.f;
                for (int j = 0; j < 100; ++j) r += inRel[tt * 100 + j] * inWr[o * 100 + j];
                r += inWrb[o];
                s += r * inAw[4 + o];
            }
            consts[tt] = s;
        }
    }
    if (t == 1) {
        for (int tt = 0; tt < 2; ++tt) {
            float s = 0.f;
            for (int o = 0; o < 2; ++o) {
                float r = 0.f;
                for (int j = 0; j < 100; ++j) r += outRel[tt * 100 + j] * outWr[o * 100 + j];
                r += outWrb[o];
                s += r * outAw[4 + o];
            }
            consts[2 + tt] = s;
        }
    }
}

// ---------- dout=2 heads: h, residual base, attention scalars (wave per node) ----------
__global__ __launch_bounds__(256) void small_heads_kernel(
    const float* __restrict__ x,
    const float* inWw,  const float* inWb,  const float* inAw,  const float* inResW,  const float* inResB,
    const float* outWw, const float* outWb, const float* outAw, const float* outResW, const float* outResB,
    float* hIn, float* hOut, float* aggrIn, float* aggrOut,
    float* sdstIn, float* ssrcIn, float* sdstOut, float* ssrcOut) {
    const int node = blockIdx.x * 8 + (threadIdx.x >> 5);
    const int lane = threadIdx.x & 31;
    const float4 x4 = *(const float4*)(x + (size_t)node * 128 + lane * 4);
    const float* rows[8] = { inWw, inWw + 128, outWw, outWw + 128,
                             inResW, inResW + 128, outResW, outResW + 128 };
    float s[8];
#pragma unroll
    for (int i = 0; i < 8; ++i) {
        float4 w4 = *(const float4*)(rows[i] + lane * 4);
        s[i] = wredsum(x4.x * w4.x + x4.y * w4.y + x4.z * w4.z + x4.w * w4.w);
    }
    if (lane == 0) {
        float hi0 = s[0] + inWb[0],  hi1 = s[1] + inWb[1];
        float ho0 = s[2] + outWb[0], ho1 = s[3] + outWb[1];
        hIn[2 * node] = hi0;  hIn[2 * node + 1] = hi1;
        hOut[2 * node] = ho0; hOut[2 * node + 1] = ho1;
        aggrIn[2 * node]  = s[4] + inResB[0];  aggrIn[2 * node + 1]  = s[5] + inResB[1];
        aggrOut[2 * node] = s[6] + outResB[0]; aggrOut[2 * node + 1] = s[7] + outResB[1];
        sdstIn[node]  = hi0 * inAw[0]  + hi1 * inAw[1];
        ssrcIn[node]  = hi0 * inAw[2]  + hi1 * inAw[3];
        sdstOut[node] = ho0 * outAw[0] + ho1 * outAw[1];
        ssrcOut[node] = ho0 * outAw[2] + ho1 * outAw[3];
    }
}

// ---------- env attention scalars from h_env (wave per node) ----------
__global__ __launch_bounds__(256) void env_scores_kernel(
    const float* __restrict__ H, const float* envAw, float* sdst, float* ssrc) {
    const int node = blockIdx.x * 8 + (threadIdx.x >> 5);
    const int lane = threadIdx.x & 31;
    const float4 h4 = *(const float4*)(H + (size_t)node * 128 + lane * 4);
    const float4 a1 = *(const float4*)(envAw + lane * 4);
    const float4 a2 = *(const float4*)(envAw + 128 + lane * 4);
    float p1 = wredsum(h4.x * a1.x + h4.y * a1.y + h4.z * a1.z + h4.w * a1.w);
    float p2 = wredsum(h4.x * a2.x + h4.y * a2.y + h4.z * a2.z + h4.w * a2.w);
    if (lane == 0) { sdst[node] = p1; ssrc[node] = p2; }
}

// ---------- dout=2 heads: edge passes ----------
__global__ void act_pass1(const int* __restrict__ ei, const int* __restrict__ et,
                          const float* sdstIn, const float* ssrcIn,
                          const float* sdstOut, const float* ssrcOut,
                          const float* consts, const float* inAb, const float* outAb,
                          unsigned* mIn, unsigned* mOut) {
    int e = blockIdx.x * blockDim.x + threadIdx.x;
    if (e >= N_EDGES) return;
    int s = ei[e], d = ei[N_EDGES + e], t = et[e];
    float li = leaky(sdstIn[d]  + ssrcIn[s]  + consts[t]     + inAb[0]);
    float lo = leaky(sdstOut[d] + ssrcOut[s] + consts[2 + t] + outAb[0]);
    atomicMax(&mIn[d],  encf(li));
    atomicMax(&mOut[d], encf(lo));
}

__global__ void act_pass2(const int* __restrict__ ei, const int* __restrict__ et,
                          const float* sdstIn, const float* ssrcIn,
                          const float* sdstOut, const float* ssrcOut,
                          const float* consts, const float* inAb, const float* outAb,
                          const unsigned* mIn, const unsigned* mOut,
                          float* eIn, float* eOut, float* denomIn, float* denomOut) {
    int e = blockIdx.x * blockDim.x + threadIdx.x;
    if (e >= N_EDGES) return;
    int s = ei[e], d = ei[N_EDGES + e], t = et[e];
    float li = leaky(sdstIn[d]  + ssrcIn[s]  + consts[t]     + inAb[0]);
    float lo = leaky(sdstOut[d] + ssrcOut[s] + consts[2 + t] + outAb[0]);
    float vi = __expf(0.f); // placeholder to keep fast path out; use expf below
    (void)vi;
    float ein = expf(li - decf(mIn[d]));
    float eon = expf(lo - decf(mOut[d]));
    eIn[e] = ein; eOut[e] = eon;
    atomicAdd(&denomIn[d],  ein);
    atomicAdd(&denomOut[d], eon);
}

__global__ void act_pass3(const int* __restrict__ ei,
                          const float* eIn, const float* eOut,
                          const float* denomIn, const float* denomOut,
                          const float* hIn, const float* hOut,
                          float* aggrIn, float* aggrOut) {
    int e = blockIdx.x * blockDim.x + threadIdx.x;
    if (e >= N_EDGES) return;
    int s = ei[e], d = ei[N_EDGES + e];
    float ai = eIn[e]  / fmaxf(denomIn[d],  1e-16f);
    float ao = eOut[e] / fmaxf(denomOut[d], 1e-16f);
    atomicAdd(&aggrIn[2 * d],      hIn[2 * s]      * ai);
    atomicAdd(&aggrIn[2 * d + 1],  hIn[2 * s + 1]  * ai);
    atomicAdd(&aggrOut[2 * d],     hOut[2 * s]     * ao);
    atomicAdd(&aggrOut[2 * d + 1], hOut[2 * s + 1] * ao);
}

// ---------- Gumbel-hard forward: keep = (logit0+g0 >= logit1+g1) ----------
__global__ void gumbel_kernel(const float* aggrIn, const float* aggrOut,
                              const float* g_in, const float* g_out,
                              int* keepIn, int* keepOut) {
    int n = blockIdx.x * blockDim.x + threadIdx.x;
    if (n >= N_NODES) return;
    keepIn[n]  = ((aggrIn[2 * n]  + g_in[2 * n])  >= (aggrIn[2 * n + 1]  + g_in[2 * n + 1]))  ? 1 : 0;
    keepOut[n] = ((aggrOut[2 * n] + g_out[2 * n]) >= (aggrOut[2 * n + 1] + g_out[2 * n + 1])) ? 1 : 0;
}

// ---------- env layer edge passes (masked) ----------
__global__ void env_pass1(const int* __restrict__ ei, const int* __restrict__ et,
                          const int* keepIn, const int* keepOut,
                          const float* sdstE, const float* ssrcE,
                          const float* consts, const float* envAb, unsigned* mEnv) {
    int e = blockIdx.x * blockDim.x + threadIdx.x;
    if (e >= N_EDGES) return;
    int s = ei[e], d = ei[N_EDGES + e], t = et[e];
    if (keepIn[d] && keepOut[s]) {
        float l = leaky(sdstE[d] + ssrcE[s] + consts[4 + t] + envAb[0]);
        atomicMax(&mEnv[d], encf(l));
    }
}

__global__ void env_pass2(const int* __restrict__ ei, const int* __restrict__ et,
                          const int* keepIn, const int* keepOut,
                          const float* sdstE, const float* ssrcE,
                          const float* consts, const float* envAb,
                          const unsigned* mEnv, float* eEnv, float* denomEnv) {
    int e = blockIdx.x * blockDim.x + threadIdx.x;
    if (e >= N_EDGES) return;
    int s = ei[e], d = ei[N_EDGES + e], t = et[e];
    if (keepIn[d] && keepOut[s]) {
        float l  = leaky(sdstE[d] + ssrcE[s] + consts[4 + t] + envAb[0]);
        float ee = expf(l - decf(mEnv[d]));
        eEnv[e] = ee;
        atomicAdd(&denomEnv[d], ee);
    } else {
        eEnv[e] = 0.f;
    }
}

// 128 threads per edge scatter alpha*h_env[src] into the pre-initialized output.
__global__ __launch_bounds__(256) void env_pass3(const int* __restrict__ ei,
                                                 const float* eEnv, const float* denomEnv,
                                                 const float* __restrict__ hEnv,
                                                 float* __restrict__ out) {
    int e = blockIdx.x * 2 + (threadIdx.x >> 7);
    int t = threadIdx.x & 127;
    float ee = eEnv[e];
    if (ee != 0.f) {
        int s = ei[e], d = ei[N_EDGES + e];
        float alpha = ee / fmaxf(denomEnv[d], 1e-16f);
        atomicAdd(&out[(size_t)d * 128 + t], alpha * hEnv[(size_t)s * 128 + t]);
    }
}

// ---------- host side ----------
extern "C" void kernel_launch(void* const* d_in, const int* in_sizes, int n_in,
                              void* d_out, int out_size, void* d_ws, size_t ws_size,
                              hipStream_t stream) {
    const float* x     = (const float*)d_in[0];
    const int*   ei    = (const int*)d_in[1];
    const int*   et    = (const int*)d_in[2];
    const float* g_in  = (const float*)d_in[3];
    const float* g_out = (const float*)d_in[4];
    // inA params: 5..13 ; outA: 14..22 ; env: 23..31
    const float* inWw   = (const float*)d_in[5];
    const float* inWb   = (const float*)d_in[6];
    const float* inWr   = (const float*)d_in[7];
    const float* inWrb  = (const float*)d_in[8];
    const float* inAw   = (const float*)d_in[9];
    const float* inAb   = (const float*)d_in[10];
    const float* inResW = (const float*)d_in[11];
    const float* inResB = (const float*)d_in[12];
    const float* inRel  = (const float*)d_in[13];
    const float* outWw   = (const float*)d_in[14];
    const float* outWb   = (const float*)d_in[15];
    const float* outWr   = (const float*)d_in[16];
    const float* outWrb  = (const float*)d_in[17];
    const float* outAw   = (const float*)d_in[18];
    const float* outAb   = (const float*)d_in[19];
    const float* outResW = (const float*)d_in[20];
    const float* outResB = (const float*)d_in[21];
    const float* outRel  = (const float*)d_in[22];
    const float* envWw   = (const float*)d_in[23];
    const float* envWb   = (const float*)d_in[24];
    const float* envWr   = (const float*)d_in[25];
    const float* envWrb  = (const float*)d_in[26];
    const float* envAw   = (const float*)d_in[27];
    const float* envAb   = (const float*)d_in[28];
    const float* envResW = (const float*)d_in[29];
    const float* envResB = (const float*)d_in[30];
    const float* envRel  = (const float*)d_in[31];

    char* ws = (char*)d_ws;
    size_t off = 0;
    auto alloc = [&](size_t bytes) -> void* {
        void* p = ws + off;
        off = (off + bytes + 255) & ~(size_t)255;
        return p;
    };
    const size_t Nf = (size_t)N_NODES * 4, Ef = (size_t)N_EDGES * 4;
    float* hEnv    = (float*)alloc((size_t)N_NODES * 128 * 4);
    float* sdstE   = (float*)alloc(Nf);
    float* ssrcE   = (float*)alloc(Nf);
    float* sdstIn  = (float*)alloc(Nf);
    float* ssrcIn  = (float*)alloc(Nf);
    float* sdstOut = (float*)alloc(Nf);
    float* ssrcOut = (float*)alloc(Nf);
    float* hIn     = (float*)alloc(Nf * 2);
    float* hOut    = (float*)alloc(Nf * 2);
    float* aggrIn  = (float*)alloc(Nf * 2);
    float* aggrOut = (float*)alloc(Nf * 2);
    int*   keepIn  = (int*)alloc(Nf);
    int*   keepOut = (int*)alloc(Nf);
    float* eIn     = (float*)alloc(Ef);
    float* eOut    = (float*)alloc(Ef);
    float* eEnv    = (float*)alloc(Ef);
    float* consts  = (float*)alloc(64);
    size_t zstart = off;
    unsigned* mIn      = (unsigned*)alloc(Nf);
    unsigned* mOut     = (unsigned*)alloc(Nf);
    unsigned* mEnv     = (unsigned*)alloc(Nf);
    float*    denomIn  = (float*)alloc(Nf);
    float*    denomOut = (float*)alloc(Nf);
    float*    denomEnv = (float*)alloc(Nf);
    size_t zbytes = off - zstart;

    float* outF = (float*)d_out;

    hipMemsetAsync(ws + zstart, 0, zbytes, stream);

    prep_rel_kernel<<<1, 128, 0, stream>>>(inWr, inWrb, inAw, inRel,
                                           outWr, outWrb, outAw, outRel,
                                           envWr, envWrb, envAw, envRel, consts);

    // env h = x@W.T + b ; output base = x@Wres.T + bres (into d_out)
    gemm128_kernel<<<N_NODES / 16, 256, 0, stream>>>(x, envWw, envWb, hEnv);
    gemm128_kernel<<<N_NODES / 16, 256, 0, stream>>>(x, envResW, envResB, outF);

    small_heads_kernel<<<N_NODES / 8, 256, 0, stream>>>(
        x, inWw, inWb, inAw, inResW, inResB,
        outWw, outWb, outAw, outResW, outResB,
        hIn, hOut, aggrIn, aggrOut, sdstIn, ssrcIn, sdstOut, ssrcOut);

    env_scores_kernel<<<N_NODES / 8, 256, 0, stream>>>(hEnv, envAw, sdstE, ssrcE);

    const int eb = 256, eg = N_EDGES / 256;
    act_pass1<<<eg, eb, 0, stream>>>(ei, et, sdstIn, ssrcIn, sdstOut, ssrcOut,
                                     consts, inAb, outAb, mIn, mOut);
    act_pass2<<<eg, eb, 0, stream>>>(ei, et, sdstIn, ssrcIn, sdstOut, ssrcOut,
                                     consts, inAb, outAb, mIn, mOut,
                                     eIn, eOut, denomIn, denomOut);
    act_pass3<<<eg, eb, 0, stream>>>(ei, eIn, eOut, denomIn, denomOut,
                                     hIn, hOut, aggrIn, aggrOut);

    gumbel_kernel<<<(N_NODES + 255) / 256, 256, 0, stream>>>(aggrIn, aggrOut,
                                                             g_in, g_out, keepIn, keepOut);

    env_pass1<<<eg, eb, 0, stream>>>(ei, et, keepIn, keepOut, sdstE, ssrcE,
                                     consts, envAb, mEnv);
    env_pass2<<<eg, eb, 0, stream>>>(ei, et, keepIn, keepOut, sdstE, ssrcE,
                                     consts, envAb, mEnv, eEnv, denomEnv);
    env_pass3<<<N_EDGES / 2, 256, 0, stream>>>(ei, eEnv, denomEnv, hEnv, outF);
}